// VectorizedQuantumCircuit_6992206758571
// MI455X (gfx1250) — compile-verified
//
#include <hip/hip_runtime.h>

// ---------------------------------------------------------------------------
// 12-qubit batched state-vector simulator for MI455X (gfx1250, wave32).
// RY + CNOT circuit on |0> is purely REAL -> fp32 state (16KB per element).
// 16 batch states resident per workgroup in WGP LDS (256.4KB of 320KB).
// Epilogue probs @ SIGNS via V_WMMA_F32_16X16X4_F32, signs built by bit tricks.
// ---------------------------------------------------------------------------

#define NQ      12
#define DIM     4096            // 2^12
#define BATCH   1024
#define BPG     16              // batch elements per workgroup
#define BLOCK   512             // 16 wave32 waves
#define STRIDE  (DIM + 4)       // 4100 mod 64 = 4 -> conflict-free column reads
#define CS_OFF  (BPG * STRIDE)  // offset (floats) of small cos/sin buffers
#define LDS_FLOATS (CS_OFF + 2 * BPG)

typedef __attribute__((ext_vector_type(2))) float v2f;
typedef __attribute__((ext_vector_type(8))) float v8f;

// RY(q) butterfly sweep over all BPG states. cb/sb hold per-batch cos/sin.
// Outer loop over batch rows hoists c/s/base; inner index is just j*512+t.
__device__ __forceinline__ void ry_sweep(float* __restrict__ st,
                                         const float* __restrict__ cb,
                                         const float* __restrict__ sb,
                                         int q, int t) {
  const int mask = (1 << q) - 1;
  for (int b = 0; b < BPG; ++b) {
    const float c = cb[b], s = sb[b];
    const int base = b * STRIDE;
#pragma unroll
    for (int j = 0; j < (DIM / 2) / BLOCK; ++j) {        // 4 chunks of 512 pairs
      const int p = j * BLOCK + t;                       // pair id in [0,2048)
      const int pos = ((p >> q) << (q + 1)) | (p & mask);// insert 0 at bit q
      const int i0 = base + pos;
      const int i1 = i0 + (1 << q);
      const float a0 = st[i0], a1 = st[i1];
      st[i0] = c * a0 - s * a1;
      st[i1] = s * a0 + c * a1;
    }
  }
}

// CNOT(ctrl,tgt): swap amplitudes (ctrl=1,tgt=0) <-> (ctrl=1,tgt=1), in place.
__device__ __forceinline__ void cnot_sweep(float* __restrict__ st,
                                           int ctrl, int tgt, int t) {
  const int lo = ctrl < tgt ? ctrl : tgt;
  const int hi = ctrl < tgt ? tgt : ctrl;
  const int mlo = (1 << lo) - 1, mhi = (1 << hi) - 1;
  for (int b = 0; b < BPG; ++b) {
    const int base = b * STRIDE;
#pragma unroll
    for (int j = 0; j < (DIM / 4) / BLOCK; ++j) {        // 2 chunks of 512 swaps
      const int m = j * BLOCK + t;                       // free-bit id in [0,1024)
      const int tmp = ((m >> lo) << (lo + 1)) | (m & mlo);    // insert 0 at lo
      const int i   = ((tmp >> hi) << (hi + 1)) | (tmp & mhi);// insert 0 at hi
      const int ia = base + (i | (1 << ctrl));                // ctrl=1, tgt=0
      const int ib = ia | (1 << tgt);                         // ctrl=1, tgt=1
      const float va = st[ia], vb = st[ib];
      st[ia] = vb;
      st[ib] = va;
    }
  }
}

__global__ __launch_bounds__(BLOCK)
void qsim_kernel(const float* __restrict__ x, const float* __restrict__ thetas,
                 float* __restrict__ out) {
  extern __shared__ float lds[];
  float* st   = lds;             // BPG padded state rows
  float* cbuf = lds + CS_OFF;    // per-batch cos
  float* sbuf = cbuf + BPG;      // per-batch sin

  const int t = threadIdx.x;
  const int blockStart = blockIdx.x * BPG;

  // ---- init: |0...0> per batch element --------------------------------
  for (int i = t; i < BPG * STRIDE; i += BLOCK) st[i] = 0.f;
  __syncthreads();
  if (t < BPG) st[t * STRIDE] = 1.f;
  __syncthreads();

  // ---- angle encoding: RY(x[b,q]) on each qubit -----------------------
  for (int q = 0; q < NQ; ++q) {
    if (t < BPG) {
      const float ang = 0.5f * x[(blockStart + t) * NQ + q];
      cbuf[t] = cosf(ang);
      sbuf[t] = sinf(ang);
    }
    __syncthreads();
    ry_sweep(st, cbuf, sbuf, q, t);
    __syncthreads();
  }

  // ---- variational layers: RY(theta) per qubit + cyclic CNOT ring -----
  for (int layer = 0; layer < 2; ++layer) {
    for (int q = 0; q < NQ; ++q) {
      if (t < BPG) {
        const float ang = 0.5f * thetas[layer * NQ + q];  // uniform over batch
        cbuf[t] = cosf(ang);
        sbuf[t] = sinf(ang);
      }
      __syncthreads();
      ry_sweep(st, cbuf, sbuf, q, t);
      __syncthreads();
    }
    for (int q = 0; q < NQ; ++q) {
      cnot_sweep(st, q, (q + 1) % NQ, t);
      __syncthreads();
    }
  }

  // ---- epilogue: out[b,q] = sum_s state[b,s]^2 * (1 - 2*bit_q(s)) -----
  // WMMA f32 16x16x4: A = probs (16 batches x 4 states), B = +/-1 signs.
  // Each of the 16 waves owns a disjoint K-chunk of 64 WMMA steps.
  const int lane = t & 31;
  const int wave = t >> 5;                 // 0..15
  const int m  = lane & 15;                // A row (batch) / B col (qubit)
  const int kb = (lane >> 4) * 2;          // K sub-offset per ISA A/B layout
  const int ktBeg = wave * ((DIM / 4) / 16);
  const int ktEnd = ktBeg + ((DIM / 4) / 16);

  // Branch-free sign construction: sign(k,q) = as_float((bit_q(k)<<31)|1.0f),
  // with columns q>=12 forced to +0.0f via zeroed per-lane constants.
  const unsigned sh    = 31u - (unsigned)m;
  const unsigned smask = (m < NQ) ? 0x80000000u : 0u;
  const unsigned obase = (m < NQ) ? 0x3f800000u : 0u;
  const int rowBase = m * STRIDE;

  v8f acc = {};
  for (int kt = ktBeg; kt < ktEnd; ++kt) {
    const int k0 = kt * 4 + kb;
    const float p0 = st[rowBase + k0];
    const float p1 = st[rowBase + k0 + 1];
    v2f a;
    a.x = p0 * p0;                         // probs on the fly (state is real)
    a.y = p1 * p1;
    v2f bv;
    bv.x = __uint_as_float((((unsigned)k0 << sh) & smask) | obase);
    bv.y = __uint_as_float((((unsigned)(k0 + 1) << sh) & smask) | obase);
    acc = __builtin_amdgcn_wmma_f32_16x16x4_f32(
        /*neg_a=*/false, a, /*neg_b=*/false, bv,
        /*c_mod=*/(short)0, acc, /*reuse_a=*/false, /*reuse_b=*/false);
  }
  __syncthreads();   // all waves done reading st -> safe to reuse as scratch

  // Spill per-wave partial C tiles (16x16 each) and sum them over waves in LDS.
  float* part = lds;  // 16 waves * 256 floats, aliases dead state region
#pragma unroll
  for (int r = 0; r < 8; ++r) {
    const int M = r + 8 * (lane >> 4);     // ISA C/D layout
    const int N = lane & 15;
    part[wave * 256 + M * 16 + N] = acc[r];
  }
  __syncthreads();

  if (t < 256) {
    const int M = t >> 4, N = t & 15;
    float sum = 0.f;
#pragma unroll
    for (int w = 0; w < 16; ++w) sum += part[w * 256 + t];
    if (N < NQ) out[(blockStart + M) * NQ + N] = sum;
  }
}

extern "C" void kernel_launch(void* const* d_in, const int* in_sizes, int n_in,
                              void* d_out, int out_size, void* d_ws, size_t ws_size,
                              hipStream_t stream) {
  (void)in_sizes; (void)n_in; (void)out_size; (void)d_ws; (void)ws_size;
  const float* x      = (const float*)d_in[0];   // (1024, 12) fp32
  const float* thetas = (const float*)d_in[1];   // (2, 12)    fp32
  float* out          = (float*)d_out;           // (1024, 12) fp32

  const size_t ldsBytes = (size_t)LDS_FLOATS * sizeof(float);  // ~256.4 KB
  qsim_kernel<<<dim3(BATCH / BPG), dim3(BLOCK), ldsBytes, stream>>>(x, thetas, out);
}